// ParabolicIntegrate_2d_34797825032619
// MI455X (gfx1250) — compile-verified
//
#include <hip/hip_runtime.h>

typedef __attribute__((ext_vector_type(2))) float v2f;
typedef __attribute__((ext_vector_type(8))) float v8f;

#define Bsz 8
#define Tsz 50
#define PLANE 4096          // 64*64
#define DTc 0.1f
#define JCc 0.15915494309189535f   // 1/(2*pi)

// lap_I = identity + 0.1 * [[.25,.5,.25],[.5,-3,.5],[.25,.5,.25]] (circular)
#define C0 0.7f
#define C1 0.05f
#define C2 0.025f

// Circular central-difference matrix (DXM == DYM, dx=dy=1):
// D[r][c] = +0.5 if c == r-1 (mod 64), -0.5 if c == r+1 (mod 64)
__device__ __forceinline__ float diffm(int r, int c) {
  if (c == ((r + 63) & 63)) return 0.5f;
  if (c == ((r + 1) & 63)) return -0.5f;
  return 0.f;
}

// ---------------------------------------------------------------------------
// Kernel A: f_I scan (one block per b; 64x64 plane lives in LDS, loop over T)
// also computes f_j[b,t] = sum(plane) / (2*pi). Prefetches next W plane to
// hide global latency on the serial dependence chain.
// ---------------------------------------------------------------------------
__global__ __launch_bounds__(256)
void integrate_fI_kernel(const float* __restrict__ W,
                         float* __restrict__ fI,
                         float* __restrict__ fj) {
  __shared__ float plane[64][65];
  __shared__ float red[256];
  const int b = blockIdx.x;
  const int tid = threadIdx.x;

  for (int i = 0; i < 16; ++i) {
    int idx = tid + 256 * i;
    plane[idx >> 6][idx & 63] = 0.f;
    fI[(size_t)(b * Tsz + 0) * PLANE + idx] = 0.f;
  }
  if (tid == 0) fj[b * Tsz + 0] = 0.f;
  __syncthreads();

  for (int t = 1; t < Tsz; ++t) {
    const float* Wt = W + (size_t)(b * Tsz + t) * PLANE;
    float* Ft = fI + (size_t)(b * Tsz + t) * PLANE;
    if (t + 1 < Tsz) {
      // pull next step's source plane toward the WGP while we compute
      __builtin_prefetch(Wt + PLANE + tid * 16, 0, 0);
    }
    float nv[16];
    float psum = 0.f;
    for (int i = 0; i < 16; ++i) {
      int idx = tid + 256 * i;
      int x = idx >> 6, y = idx & 63;
      int xm = (x + 63) & 63, xp = (x + 1) & 63;
      int ym = (y + 63) & 63, yp = (y + 1) & 63;
      float lap = C0 * plane[x][y]
                + C1 * (plane[xm][y] + plane[xp][y] + plane[x][ym] + plane[x][yp])
                + C2 * (plane[xm][ym] + plane[xm][yp] + plane[xp][ym] + plane[xp][yp]);
      float v = lap + Wt[idx] * DTc;
      nv[i] = v;
      psum += v;
    }
    __syncthreads();
    for (int i = 0; i < 16; ++i) {
      int idx = tid + 256 * i;
      plane[idx >> 6][idx & 63] = nv[i];
      Ft[idx] = nv[i];
    }
    red[tid] = psum;
    __syncthreads();
    for (int s = 128; s > 0; s >>= 1) {
      if (tid < s) red[tid] += red[tid + s];
      __syncthreads();
    }
    if (tid == 0) fj[b * Tsz + t] = red[0] * JCc;
  }
}

// ---------------------------------------------------------------------------
// Kernel B: f_dx = DXM^T x fI, f_dy = fI x DYM via V_WMMA_F32_16X16X4_F32.
// One block per (b,t); fI plane staged in padded LDS; difference-matrix
// fragments synthesized analytically in registers. K-loops fully unrolled so
// LDS fragment loads pipeline under the WMMA chain.
// ---------------------------------------------------------------------------
__global__ __launch_bounds__(256)
void deriv_wmma_kernel(const float* __restrict__ fI,
                       float* __restrict__ fdx,
                       float* __restrict__ fdy) {
  __shared__ float sI[64][65];
  const int bt = blockIdx.x;
  const int tid = threadIdx.x;
  const float* src = fI + (size_t)bt * PLANE;
  for (int i = 0; i < 16; ++i) {
    int idx = tid + 256 * i;
    sI[idx >> 6][idx & 63] = src[idx];
  }
  __syncthreads();

  const int wave = tid >> 5;
  const int lane = tid & 31;
  const int h = lane >> 4;     // lane half: K offset of 2, D row offset of 8
  const int lm = lane & 15;

  // f_dx[n,y] = sum_x DXM[x,n] * fI[x,y]  (A[m][k] = DXM[k][m], B = fI)
  for (int tt = wave; tt < 16; tt += 8) {
    const int m0 = (tt >> 2) * 16;
    const int n0 = (tt & 3) * 16;
    v8f acc = {};
#pragma unroll
    for (int k0 = 0; k0 < 64; k0 += 4) {
      const int k = k0 + 2 * h;
      v2f a, bb;
      a.x = diffm(k, m0 + lm);
      a.y = diffm(k + 1, m0 + lm);
      bb.x = sI[k][n0 + lm];
      bb.y = sI[k + 1][n0 + lm];
      acc = __builtin_amdgcn_wmma_f32_16x16x4_f32(
          false, a, false, bb, (short)0, acc, false, false);
    }
    float* dst = fdx + (size_t)bt * PLANE;
#pragma unroll
    for (int r = 0; r < 8; ++r)
      dst[(m0 + r + 8 * h) * 64 + (n0 + lm)] = acc[r];
  }

  // f_dy[x,n] = sum_y fI[x,y] * DYM[y,n]  (A = fI, B[k][n] = DYM[k][n])
  for (int tt = wave; tt < 16; tt += 8) {
    const int m0 = (tt >> 2) * 16;
    const int n0 = (tt & 3) * 16;
    v8f acc = {};
#pragma unroll
    for (int k0 = 0; k0 < 64; k0 += 4) {
      const int k = k0 + 2 * h;
      v2f a, bb;
      a.x = sI[m0 + lm][k];
      a.y = sI[m0 + lm][k + 1];
      bb.x = diffm(k, n0 + lm);
      bb.y = diffm(k + 1, n0 + lm);
      acc = __builtin_amdgcn_wmma_f32_16x16x4_f32(
          false, a, false, bb, (short)0, acc, false, false);
    }
    float* dst = fdy + (size_t)bt * PLANE;
#pragma unroll
    for (int r = 0; r < 8; ++r)
      dst[(m0 + r + 8 * h) * 64 + (n0 + lm)] = acc[r];
  }
}

// ---------------------------------------------------------------------------
// Kernel C: six independent stencil scans -> features 5..10, written directly
// into the stacked output. One block per (b, chain) = 48 blocks.
// ---------------------------------------------------------------------------
__global__ __launch_bounds__(256)
void integrate_feats_kernel(const float* __restrict__ fI,
                            const float* __restrict__ fdx,
                            const float* __restrict__ fdy,
                            const float* __restrict__ fj,
                            float* __restrict__ out) {
  __shared__ float plane[64][65];
  const int b = blockIdx.x / 6;
  const int chain = blockIdx.x % 6;
  const int feat = 5 + chain;
  const int tid = threadIdx.x;

  for (int i = 0; i < 16; ++i) {
    int idx = tid + 256 * i;
    plane[idx >> 6][idx & 63] = 0.f;
    out[(size_t)((b * Tsz + 0) * PLANE + idx) * 11 + feat] = 0.f;
  }
  __syncthreads();

  for (int t = 1; t < Tsz; ++t) {
    const size_t base = (size_t)(b * Tsz + t) * PLANE;
    const float jv = fj[b * Tsz + t];
    float nv[16];
    for (int i = 0; i < 16; ++i) {
      int idx = tid + 256 * i;
      int x = idx >> 6, y = idx & 63;
      int xm = (x + 63) & 63, xp = (x + 1) & 63;
      int ym = (y + 63) & 63, yp = (y + 1) & 63;
      float lap = C0 * plane[x][y]
                + C1 * (plane[xm][y] + plane[xp][y] + plane[x][ym] + plane[x][yp])
                + C2 * (plane[xm][ym] + plane[xm][yp] + plane[xp][ym] + plane[xp][yp]);
      float s;
      switch (chain) {
        case 0: { float v = fI[base + idx];  s = v * v * DTc; } break;       // I[I^2]
        case 1: { float v = fI[base + idx];  s = v * v * v * DTc; } break;   // I[I^3]
        case 2: { float v = fdx[base + idx]; s = v * v * DTc; } break;       // I[dx^2]
        case 3: { s = fI[base + idx] * fdx[base + idx] * DTc; } break;       // I[I*dx]
        case 4: { s = fI[base + idx] * fdy[base + idx] * DTc; } break;       // I[I*dy]
        default:{ s = jv * jv * DTc; } break;                                // I[J^2]
      }
      nv[i] = lap + s;
    }
    __syncthreads();
    for (int i = 0; i < 16; ++i) {
      int idx = tid + 256 * i;
      plane[idx >> 6][idx & 63] = nv[i];
      out[(base + idx) * 11 + feat] = nv[i];
    }
    __syncthreads();
  }
}

// ---------------------------------------------------------------------------
// Kernel D: write features 0..4 (f_xi, f_I, f_dx, f_dy, f_j broadcast)
// ---------------------------------------------------------------------------
__global__ __launch_bounds__(256)
void stack_base_kernel(const float* __restrict__ W,
                       const float* __restrict__ fI,
                       const float* __restrict__ fdx,
                       const float* __restrict__ fdy,
                       const float* __restrict__ fj,
                       float* __restrict__ out) {
  const size_t g = (size_t)blockIdx.x * 256 + threadIdx.x;
  const size_t N = (size_t)Bsz * Tsz * PLANE;
  if (g >= N) return;
  const int bt = (int)(g >> 12);  // PLANE == 4096
  const size_t o = g * 11;
  out[o + 0] = W[g];
  out[o + 1] = fI[g];
  out[o + 2] = fdx[g];
  out[o + 3] = fdy[g];
  out[o + 4] = fj[bt];
}

// ---------------------------------------------------------------------------
extern "C" void kernel_launch(void* const* d_in, const int* in_sizes, int n_in,
                              void* d_out, int out_size, void* d_ws, size_t ws_size,
                              hipStream_t stream) {
  const float* W = (const float*)d_in[0];
  float* out = (float*)d_out;
  float* ws = (float*)d_ws;

  const size_t NP = (size_t)Bsz * Tsz * PLANE;  // 1,638,400 floats
  float* fI  = ws;                              // [B,T,X,Y]
  float* fdx = ws + NP;                         // [B,T,X,Y]
  float* fdy = ws + 2 * NP;                     // [B,T,X,Y]
  float* fj  = ws + 3 * NP;                     // [B,T] scalars
  // total ws use: ~19.7 MB

  integrate_fI_kernel<<<Bsz, 256, 0, stream>>>(W, fI, fj);
  deriv_wmma_kernel<<<Bsz * Tsz, 256, 0, stream>>>(fI, fdx, fdy);
  integrate_feats_kernel<<<Bsz * 6, 256, 0, stream>>>(fI, fdx, fdy, fj, out);
  stack_base_kernel<<<(int)((NP + 255) / 256), 256, 0, stream>>>(W, fI, fdx, fdy, fj, out);
}